// Attention_11665131176279
// MI455X (gfx1250) — compile-verified
//
#include <hip/hip_runtime.h>

#define B_    64
#define T_    2048
#define ENC   512
#define ATT   128
#define RNN   1024
#define NF    32
#define KS    31
#define WIN_  32
#define WSPAN 65      // 2*WIN+1 window positions
#define PADW  15      // (KS-1)/2

typedef __attribute__((ext_vector_type(2))) float v2f;
typedef __attribute__((ext_vector_type(8))) float v8f;

// -------------------------------------------------------------------------
// Kernel 1: Q = hidden[64,1024] @ query_w.T[1024,128] via V_WMMA_F32_16X16X4_F32
// One workgroup, 32 waves; wave (mt,nt) owns 16x16 tile of Q[64,128].
// A layout (16x4 f32): lane m%16 holds row m; k = vgpr + 2*(lane>=16)
// B layout (4x16 f32): row k striped across lanes; k = vgpr + 2*(lane>=16)
// D layout: row = vgpr + 8*(lane>=16), col = lane&15
// -------------------------------------------------------------------------
__global__ __launch_bounds__(1024) void query_proj_wmma(
    const float* __restrict__ hidden,    // [64][1024]
    const float* __restrict__ query_w,   // [128][1024]  (B[k][n] = query_w[n][k])
    float* __restrict__ qout)            // [64][128]
{
    const int tid  = threadIdx.x;
    const int wave = tid >> 5;
    const int lane = tid & 31;
    const int half = lane >> 4;
    const int l16  = lane & 15;
    const int mt   = wave >> 3;          // 0..3
    const int nt   = wave & 7;           // 0..7
    const int m    = mt * 16 + l16;
    const int n    = nt * 16 + l16;

    const float* __restrict__ arow = hidden  + (size_t)m * RNN;
    const float* __restrict__ brow = query_w + (size_t)n * RNN;

    v8f c = {};
#pragma unroll 4
    for (int k0 = 0; k0 < RNN; k0 += 4) {
        v2f a, b;
        a.x = arow[k0 + 2 * half];
        a.y = arow[k0 + 2 * half + 1];
        b.x = brow[k0 + 2 * half];
        b.y = brow[k0 + 2 * half + 1];
        c = __builtin_amdgcn_wmma_f32_16x16x4_f32(false, a, false, b,
                                                  (short)0, c, false, false);
    }
#pragma unroll
    for (int v = 0; v < 8; ++v) {
        int row = mt * 16 + v + 8 * half;
        qout[(size_t)row * ATT + nt * 16 + l16] = c[v];
    }
}

// -------------------------------------------------------------------------
// Kernel 2: per-batch windowed attention. 1 block per batch, 256 threads.
// Only the 65-position window contributes (outside: weight == 0 exactly).
// -------------------------------------------------------------------------
__global__ __launch_bounds__(256) void windowed_attention(
    const float* __restrict__ memory,    // [B][T][512]
    const float* __restrict__ pmem,      // [B][T][128]
    const float* __restrict__ awcat,     // [B][2][T]
    const unsigned char* __restrict__ mask, // [B][T] (bool)
    const int*   __restrict__ mlen,      // [B]
    const float* __restrict__ curpos,    // [B]
    const float* __restrict__ convw,     // [32][2][31]
    const float* __restrict__ densew,    // [128][32]
    const float* __restrict__ vw,        // [128]
    const float* __restrict__ posoff,    // [1]
    const float* __restrict__ qws,       // [B][128]  (from kernel 1)
    float* __restrict__ out_ctx,         // [B][512]
    float* __restrict__ out_w,           // [B][T]
    float* __restrict__ out_pos)         // [B]
{
    __shared__ float s_aw[2][112];       // padded conv input window (95 used)
    __shared__ float s_cw[NF][64];       // conv weights [f][c*31+k] (62 used)
    __shared__ float s_conv[80][33];     // conv output, rows 65..79 zero, pad col
    __shared__ float s_part[8][80];      // per-wave energy partials
    __shared__ float s_energy[80];
    __shared__ float s_wt[66];
    __shared__ int   s_start;

    const int b   = blockIdx.x;
    const int tid = threadIdx.x;

    // ---- window start (matches reference clamp + round-to-nearest-even) ----
    if (tid == 0) {
        float cp      = curpos[b] + posoff[0];
        float max_end = (float)(mlen[b] - 1 - WIN_);
        cp = fminf(fmaxf(cp, (float)WIN_), max_end);
        s_start = (int)rintf(fmaxf(cp - (float)WIN_, 0.0f));
    }
    __syncthreads();
    const int start = s_start;

    // ---- stage conv inputs/weights; zero conv buffer ----
    for (int p = tid; p < 2 * 95; p += 256) {
        int c = p / 95, j = p % 95;
        int g = start - PADW + j;
        float v = 0.0f;
        if (g >= 0 && g < T_) v = awcat[((size_t)b * 2 + c) * T_ + g];
        s_aw[c][j] = v;
    }
    for (int p = tid; p < NF * 2 * KS; p += 256) {
        int f = p / (2 * KS), r = p % (2 * KS);
        s_cw[f][r] = convw[p];
    }
    for (int p = tid; p < 80 * 33; p += 256) ((float*)s_conv)[p] = 0.0f;
    __syncthreads();

    // ---- location conv: 65 positions x 32 filters ----
    for (int p = tid; p < WSPAN * NF; p += 256) {
        int f = p & 31, t = p >> 5;
        float acc = 0.0f;
#pragma unroll
        for (int c = 0; c < 2; ++c)
#pragma unroll
            for (int k = 0; k < KS; ++k)
                acc += s_aw[c][t + k] * s_cw[f][c * KS + k];
        s_conv[t][f] = acc;
    }
    __syncthreads();

    // ---- dense 32->128 via WMMA f32 16x16x4, fused +q +pmem, tanh, dot(v) ----
    const int wave = tid >> 5;           // N-tile id (0..7)
    const int lane = tid & 31;
    const int half = lane >> 4;
    const int l16  = lane & 15;
    const int acol = wave * 16 + l16;    // attention dim index 0..127

    // B fragments: B[k][n] = densew[n*32 + k]; held across all M-tiles
    v2f bf[8];
#pragma unroll
    for (int ks = 0; ks < 8; ++ks) {
        int k = ks * 4 + 2 * half;
        bf[ks].x = densew[acol * NF + k];
        bf[ks].y = densew[acol * NF + k + 1];
    }
    const float qv = qws[(size_t)b * ATT + acol];
    const float vv = vw[acol];

    for (int mt = 0; mt < 5; ++mt) {
        v8f c = {};
#pragma unroll
        for (int ks = 0; ks < 8; ++ks) {
            int k = ks * 4 + 2 * half;
            v2f a;
            a.x = s_conv[mt * 16 + l16][k];
            a.y = s_conv[mt * 16 + l16][k + 1];
            c = __builtin_amdgcn_wmma_f32_16x16x4_f32(false, a, false, bf[ks],
                                                      (short)0, c, false, false);
        }
#pragma unroll
        for (int v = 0; v < 8; ++v) {
            int i = mt * 16 + v + 8 * half;   // window position 0..79
            float val = 0.0f;
            if (i < WSPAN) {
                float x = c[v] + qv
                        + pmem[((size_t)b * T_ + start + i) * ATT + acol];
                val = tanhf(x) * vv;
            }
            // sum over the 16 lanes of this half-group (all share row i)
#pragma unroll
            for (int s = 1; s < 16; s <<= 1)
                val += __shfl_xor(val, s, 32);
            if (l16 == 0) s_part[wave][i] = val;   // deterministic, no atomics
        }
    }
    __syncthreads();

    if (tid < 80) {
        float e = 0.0f;
#pragma unroll
        for (int w8 = 0; w8 < 8; ++w8) e += s_part[w8][tid];
        s_energy[tid] = e;
    }
    __syncthreads();

    // ---- masked softmax over the 65-entry window + new_pos ----
    if (tid == 0) {
        float mx = -3.0e38f;
        for (int i = 0; i < WSPAN; ++i) {
            float e = s_energy[i];
            if (mask[(size_t)b * T_ + start + i]) e = -1.0e9f;
            s_energy[i] = e;
            mx = fmaxf(mx, e);
        }
        float sum = 0.0f;
        for (int i = 0; i < WSPAN; ++i) {
            float ex = expf(s_energy[i] - mx);
            s_wt[i] = ex;
            sum += ex;
        }
        float inv = 1.0f / sum;
        float np  = 0.0f;
        for (int i = 0; i < WSPAN; ++i) {
            float w = s_wt[i] * inv;
            s_wt[i] = w;
            np += w * (float)(start + i);
        }
        out_pos[b] = np;
    }
    __syncthreads();

    // ---- attention_weights: zero outside window, scatter inside ----
    for (int t = tid; t < T_; t += 256) {
        int i = t - start;
        float w = (i >= 0 && i < WSPAN) ? s_wt[i] : 0.0f;
        out_w[(size_t)b * T_ + t] = w;
    }

    // ---- context = w @ memory over the window (coalesced in d) ----
    for (int d = tid; d < ENC; d += 256) {
        float acc = 0.0f;
        for (int i = 0; i < WSPAN; ++i)
            acc += s_wt[i] * memory[((size_t)b * T_ + start + i) * ENC + d];
        out_ctx[(size_t)b * ENC + d] = acc;
    }
}

// -------------------------------------------------------------------------
extern "C" void kernel_launch(void* const* d_in, const int* in_sizes, int n_in,
                              void* d_out, int out_size, void* d_ws, size_t ws_size,
                              hipStream_t stream) {
    const float* hidden = (const float*)d_in[0];   // [64,1024]
    const float* memory = (const float*)d_in[1];   // [64,2048,512]
    const float* pmem   = (const float*)d_in[2];   // [64,2048,128]
    const float* awcat  = (const float*)d_in[3];   // [64,2,2048]
    const unsigned char* mask = (const unsigned char*)d_in[4]; // [64,2048] bool
    const int*   mlen   = (const int*)d_in[5];     // [64]
    const float* curpos = (const float*)d_in[6];   // [64]
    const float* convw  = (const float*)d_in[7];   // [32,2,31]
    const float* densew = (const float*)d_in[8];   // [128,32]
    const float* queryw = (const float*)d_in[9];   // [128,1024]
    const float* vw     = (const float*)d_in[10];  // [1,128]
    const float* posoff = (const float*)d_in[11];  // [1]

    float* qws     = (float*)d_ws;                 // 64*128 floats scratch
    float* out     = (float*)d_out;
    float* out_ctx = out;                          // [64][512]
    float* out_w   = out + (size_t)B_ * ENC;       // [64][2048]
    float* out_pos = out + (size_t)B_ * ENC + (size_t)B_ * T_; // [64]

    query_proj_wmma<<<1, 1024, 0, stream>>>(hidden, queryw, qws);
    windowed_attention<<<B_, 256, 0, stream>>>(memory, pmem, awcat, mask, mlen,
                                               curpos, convw, densew, vw, posoff,
                                               qws, out_ctx, out_w, out_pos);
}